// FusedSSMFFN_24739011625087
// MI455X (gfx1250) — compile-verified
//
#include <hip/hip_runtime.h>
#include <hip/hip_bf16.h>
#include <math.h>

// ---------------------------------------------------------------------------
// Fused Mamba-style SSM + FFN block for MI455X (gfx1250, wave32).
// All GEMMs run through V_WMMA_F32_16X16X4_F32 (full fp32 matrix path),
// fed from LDS-staged panels (coalesced float4 global loads, ds_load_b64
// fragment reads, bank-conflict-free padded layout).
// ---------------------------------------------------------------------------

typedef float v2f __attribute__((ext_vector_type(2)));
typedef float v8f __attribute__((ext_vector_type(8)));

#define T_LEN  2048
#define D_DIM  512
#define DS_DIM 32
#define RMS_EPS 1e-6f
#define KC 32            // K-chunk staged in LDS per iteration
#define LDP 36           // padded row length (floats): 144B = 16B-aligned, bank-spread

__device__ __forceinline__ float silu_f(float x) { return x / (1.0f + expf(-x)); }

// ------------------------------- RMSNorm -----------------------------------
__global__ void rmsnorm_kernel(const float* __restrict__ x,
                               const float* __restrict__ w,
                               float* __restrict__ xn) {
    const int row = blockIdx.x;
    const float* xr = x + row * D_DIM;
    const int tid = threadIdx.x;              // 256 threads, 2 elems each
    float v0 = xr[tid];
    float v1 = xr[tid + 256];
    float s = v0 * v0 + v1 * v1;
    #pragma unroll
    for (int off = 16; off > 0; off >>= 1) s += __shfl_xor(s, off, 32);
    __shared__ float red[8];
    const int wave = tid >> 5, lane = tid & 31;
    if (lane == 0) red[wave] = s;
    __syncthreads();
    float tot = 0.f;
    #pragma unroll
    for (int i = 0; i < 8; ++i) tot += red[i];
    const float scale = rsqrtf(tot * (1.0f / D_DIM) + RMS_EPS);
    xn[row * D_DIM + tid]       = v0 * scale * w[tid];
    xn[row * D_DIM + tid + 256] = v1 * scale * w[tid + 256];
}

// ------------------------- WMMA fp32 GEMM (LDS staged) ---------------------
// C[M,N] = epilogue(A[M,K] @ B[K,N]); 8 waves/block -> 64x32 block tile,
// one wave computes a 16x16 tile.  K stepped by 4 via V_WMMA_F32_16X16X4_F32.
// A panel (64xKC) staged row-major in LDS; B panel (KCx32) staged TRANSPOSED
// so each lane's {K=ka,K=ka+1} B fragment is one 8-byte ds_load.
enum { EPI_NONE = 0, EPI_SOFTPLUS_BIAS = 1, EPI_GELU = 2, EPI_ACCUM = 3 };

template <int EPI>
__global__ __launch_bounds__(256) void gemm_wmma_f32(
        const float* __restrict__ A, const float* __restrict__ B,
        float* __restrict__ C, const float* __restrict__ bias,
        int M, int N, int K) {
    __shared__ __align__(16) float As[64][LDP];   // [m][k]
    __shared__ __align__(16) float Bs[32][LDP];   // [n][k]  (transposed)

    const int tid  = threadIdx.x;
    const int wave = tid >> 5;
    const int lane = tid & 31;
    const int half = lane >> 4;      // 0: K={0,1}, 1: K={2,3}
    const int l    = lane & 15;
    const int bm0 = blockIdx.x * 64;
    const int bn0 = blockIdx.y * 32;
    const int m_off = (wave & 3) * 16;
    const int n_off = (wave >> 2) * 16;

    // staging coordinates (fully coalesced float4 panel loads)
    const int ar = tid >> 3;            // A row slot (0..31; +32 for second half)
    const int ac = (tid & 7) * 4;       // A col within K-chunk
    const int br = tid >> 3;            // B k-row (0..31)
    const int bc = (tid & 7) * 4;       // B n-col within 32-wide tile

    const float* aptr0 = A + (size_t)(bm0 + ar) * K + ac;
    const float* aptr1 = A + (size_t)(bm0 + ar + 32) * K + ac;
    const float* bptr  = B + (size_t)br * N + bn0 + bc;
    const size_t bstep = (size_t)KC * N;

    v8f acc = {};
    for (int k0 = 0; k0 < K; k0 += KC) {
        const float4 a0 = *reinterpret_cast<const float4*>(aptr0);
        const float4 a1 = *reinterpret_cast<const float4*>(aptr1);
        const float4 bv = *reinterpret_cast<const float4*>(bptr);
        if (k0 + KC < K) {              // prefetch next panels (global_prefetch_b8)
            __builtin_prefetch(aptr0 + KC, 0, 0);
            __builtin_prefetch(aptr1 + KC, 0, 0);
            __builtin_prefetch(bptr + bstep, 0, 0);
        }
        __syncthreads();                 // previous chunk's LDS reads done
        *reinterpret_cast<float4*>(&As[ar][ac])      = a0;
        *reinterpret_cast<float4*>(&As[ar + 32][ac]) = a1;
        Bs[bc + 0][br] = bv.x;
        Bs[bc + 1][br] = bv.y;
        Bs[bc + 2][br] = bv.z;
        Bs[bc + 3][br] = bv.w;
        __syncthreads();                 // panels visible
        #pragma unroll
        for (int kk = 0; kk < KC; kk += 4) {
            const int ka = kk + 2 * half;
            const float2 af = *reinterpret_cast<const float2*>(&As[m_off + l][ka]);
            const float2 bf = *reinterpret_cast<const float2*>(&Bs[n_off + l][ka]);
            v2f a; a.x = af.x; a.y = af.y;     // VGPR0=K(ka), VGPR1=K(ka+1)
            v2f b; b.x = bf.x; b.y = bf.y;
            acc = __builtin_amdgcn_wmma_f32_16x16x4_f32(
                false, a, false, b, (short)0, acc, false, false);
        }
        aptr0 += KC; aptr1 += KC; bptr += bstep;
    }

    const int col = bn0 + n_off + l;
    #pragma unroll
    for (int r = 0; r < 8; ++r) {
        const int row = bm0 + m_off + r + 8 * half;   // C/D layout: VGPR r, half-split M
        float v = acc[r];
        if (EPI == EPI_SOFTPLUS_BIAS) {
            v += bias[col];
            v = (v > 20.0f) ? v : log1pf(expf(v));
        } else if (EPI == EPI_GELU) {
            v = 0.5f * v * (1.0f + erff(v * 0.70710678118654752f));
        }
        float* cp = C + (size_t)row * N + col;
        if (EPI == EPI_ACCUM) *cp += v; else *cp = v;
    }
}

// -------------------- causal depthwise conv (K=4) + SiLU -------------------
__global__ void conv_silu_kernel(const float* __restrict__ xz,   // [T, 2D], xs_raw = first D cols
                                 const float* __restrict__ cw,   // [D,1,4]
                                 const float* __restrict__ cb,   // [D]
                                 float* __restrict__ xs) {       // [T, D]
    const int idx = blockIdx.x * blockDim.x + threadIdx.x;
    if (idx >= T_LEN * D_DIM) return;
    const int t = idx / D_DIM, d = idx % D_DIM;
    float acc = cb[d];
    #pragma unroll
    for (int kk = 0; kk < 4; ++kk) {
        const int tt = t - 3 + kk;
        if (tt >= 0) acc += cw[d * 4 + kk] * xz[(size_t)tt * (2 * D_DIM) + d];
    }
    xs[idx] = silu_f(acc);
}

// ------------------------------ SSM scan -----------------------------------
// One wave per channel d; lane = state index s (DS=32 == wave32).
// h[t] = exp(clip(dt*A,-20,20))*h[t-1] + dt*xs*B[t,s];  y = wave-reduce(h*C).
// Fuses the (+xs*D_param)*silu(z) epilogue.
__global__ void ssm_scan_kernel(const float* __restrict__ xs,    // [T, D]
                                const float* __restrict__ dt,    // [T, D]
                                const float* __restrict__ Bm,    // [T, DS]
                                const float* __restrict__ Cm,    // [T, DS]
                                const float* __restrict__ xz,    // [T, 2D] (z = cols D..2D)
                                const float* __restrict__ log_A, // [D]
                                const float* __restrict__ Dp,    // [D]
                                float* __restrict__ ysi) {       // [T, D]
    const int wave = threadIdx.x >> 5;          // 16 waves/block
    const int s    = threadIdx.x & 31;
    const int d    = blockIdx.x * 16 + wave;
    const float A   = -expf(log_A[d]);
    const float Dpd = Dp[d];
    float h = 0.0f;
    for (int t = 0; t < T_LEN; ++t) {
        const float dtv = dt[(size_t)t * D_DIM + d];
        const float xv  = xs[(size_t)t * D_DIM + d];
        const float Bv  = Bm[t * DS_DIM + s];
        const float Cv  = Cm[t * DS_DIM + s];
        float la = dtv * A;
        la = fminf(20.0f, fmaxf(-20.0f, la));
        h = expf(la) * h + dtv * xv * Bv;
        float contrib = h * Cv;
        #pragma unroll
        for (int off = 16; off > 0; off >>= 1) contrib += __shfl_xor(contrib, off, 32);
        if (s == 0) {
            const float zv = xz[(size_t)t * (2 * D_DIM) + D_DIM + d];
            ysi[(size_t)t * D_DIM + d] = (contrib + xv * Dpd) * silu_f(zv);
        }
    }
}

// ------------------------------ launcher -----------------------------------
extern "C" void kernel_launch(void* const* d_in, const int* in_sizes, int n_in,
                              void* d_out, int out_size, void* d_ws, size_t ws_size,
                              hipStream_t stream) {
    (void)in_sizes; (void)n_in; (void)out_size; (void)ws_size;

    const float* x       = (const float*)d_in[0];
    const float* norm_w  = (const float*)d_in[1];
    const float* W_in    = (const float*)d_in[2];
    const float* conv_w  = (const float*)d_in[3];
    const float* conv_b  = (const float*)d_in[4];
    const float* W_dt    = (const float*)d_in[5];
    const float* b_dt    = (const float*)d_in[6];
    const float* W_B     = (const float*)d_in[7];
    const float* W_C     = (const float*)d_in[8];
    const float* W_out   = (const float*)d_in[9];
    const float* log_A   = (const float*)d_in[10];
    const float* D_param = (const float*)d_in[11];
    const float* W_ffn1  = (const float*)d_in[12];
    const float* W_ffn2  = (const float*)d_in[13];
    const float* W_merge = (const float*)d_in[14];
    float* out = (float*)d_out;

    float* ws  = (float*)d_ws;
    float* xn  = ws;                              // [T, D]
    float* xz  = xn  + (size_t)T_LEN * D_DIM;     // [T, 2D]
    float* xs  = xz  + (size_t)T_LEN * 2 * D_DIM; // [T, D]
    float* dtb = xs  + (size_t)T_LEN * D_DIM;     // [T, D]
    float* Bm  = dtb + (size_t)T_LEN * D_DIM;     // [T, DS]
    float* Cm  = Bm  + (size_t)T_LEN * DS_DIM;    // [T, DS]
    float* ysi = Cm  + (size_t)T_LEN * DS_DIM;    // [T, D]
    float* t1  = ysi + (size_t)T_LEN * D_DIM;     // [T, D]  y_ssm @ W_out
    float* g   = t1  + (size_t)T_LEN * D_DIM;     // [T, 2D] gelu(xn @ W_ffn1)
    float* t2  = g   + (size_t)T_LEN * 2 * D_DIM; // [T, D]  y_ffn

    // 1) RMSNorm
    rmsnorm_kernel<<<T_LEN, 256, 0, stream>>>(x, norm_w, xn);
    // 2) xz = xn @ W_in  [2048 x 1024]
    gemm_wmma_f32<EPI_NONE><<<dim3(T_LEN / 64, (2 * D_DIM) / 32), 256, 0, stream>>>(
        xn, W_in, xz, nullptr, T_LEN, 2 * D_DIM, D_DIM);
    // 3) causal depthwise conv + SiLU -> xs
    conv_silu_kernel<<<(T_LEN * D_DIM + 255) / 256, 256, 0, stream>>>(xz, conv_w, conv_b, xs);
    // 4) dt = softplus(xs @ W_dt + b_dt)
    gemm_wmma_f32<EPI_SOFTPLUS_BIAS><<<dim3(T_LEN / 64, D_DIM / 32), 256, 0, stream>>>(
        xs, W_dt, dtb, b_dt, T_LEN, D_DIM, D_DIM);
    // 5) Bm / Cm projections [2048 x 32]
    gemm_wmma_f32<EPI_NONE><<<dim3(T_LEN / 64, DS_DIM / 32), 256, 0, stream>>>(
        xs, W_B, Bm, nullptr, T_LEN, DS_DIM, D_DIM);
    gemm_wmma_f32<EPI_NONE><<<dim3(T_LEN / 64, DS_DIM / 32), 256, 0, stream>>>(
        xs, W_C, Cm, nullptr, T_LEN, DS_DIM, D_DIM);
    // 6) SSM recurrence + (+xs*D)*silu(z) epilogue -> ysi
    ssm_scan_kernel<<<D_DIM / 16, 512, 0, stream>>>(xs, dtb, Bm, Cm, xz, log_A, D_param, ysi);
    // 7) t1 = ysi @ W_out
    gemm_wmma_f32<EPI_NONE><<<dim3(T_LEN / 64, D_DIM / 32), 256, 0, stream>>>(
        ysi, W_out, t1, nullptr, T_LEN, D_DIM, D_DIM);
    // 8) g = gelu(xn @ W_ffn1)
    gemm_wmma_f32<EPI_GELU><<<dim3(T_LEN / 64, (2 * D_DIM) / 32), 256, 0, stream>>>(
        xn, W_ffn1, g, nullptr, T_LEN, 2 * D_DIM, D_DIM);
    // 9) t2 = g @ W_ffn2
    gemm_wmma_f32<EPI_NONE><<<dim3(T_LEN / 64, D_DIM / 32), 256, 0, stream>>>(
        g, W_ffn2, t2, nullptr, T_LEN, D_DIM, 2 * D_DIM);
    // 10) out = x (residual seed), then out += t1 @ Wm_top + t2 @ Wm_bot
    hipMemcpyAsync(out, x, (size_t)T_LEN * D_DIM * sizeof(float),
                   hipMemcpyDeviceToDevice, stream);
    gemm_wmma_f32<EPI_ACCUM><<<dim3(T_LEN / 64, D_DIM / 32), 256, 0, stream>>>(
        t1, W_merge, out, nullptr, T_LEN, D_DIM, D_DIM);
    gemm_wmma_f32<EPI_ACCUM><<<dim3(T_LEN / 64, D_DIM / 32), 256, 0, stream>>>(
        t2, W_merge + (size_t)D_DIM * D_DIM, out, nullptr, T_LEN, D_DIM, D_DIM);
}